// TimeSpacePlane_49503793053728
// MI455X (gfx1250) — compile-verified
//
#include <hip/hip_runtime.h>
#include <hip/hip_bf16.h>
#include <stdint.h>

// Problem constants (from reference setup)
#define CH   32
#define RR   512          // R
#define TT   128          // T
#define HW   (1024*1024)  // H*W
#define NPIX (RR*RR)      // 262144 positions in space plane
#define SP_ELEMS ((size_t)NPIX * CH)          // packed space plane, 33.5 MB
#define COL_ELEMS (2 * CH * RR)               // 32768 floats = 128 KB

typedef float v4f __attribute__((ext_vector_type(4)));
typedef int   v4i __attribute__((vector_size(16)));

__device__ __forceinline__ int iclamp(int v, int lo, int hi) {
    return v < lo ? lo : (v > hi ? hi : v);
}

// ---------------------------------------------------------------------------
// Kernel 1a: repack space_planes[idx] from (C, R, R) to channels-last (R*R, C)
// LDS-tiled 32x32 transpose: both global read and write are 128B coalesced.
// ---------------------------------------------------------------------------
__global__ __launch_bounds__(256) void pack_space(const float* __restrict__ sp_all,
                                                  const int* __restrict__ idxp,
                                                  float* __restrict__ outp) {
    const float* img = sp_all + (size_t)(*idxp) * CH * NPIX;
    __shared__ float tile[32 * 33];   // +1 pad -> conflict-free (gcd(33,64)=1)
    const int n0 = blockIdx.x * 32;   // 32 consecutive spatial positions
    const int t = threadIdx.x;
#pragma unroll
    for (int k = 0; k < 4; ++k) {
        int lin = t + k * 256;
        int c = lin >> 5, noff = lin & 31;      // wave: fixed c, noff 0..31 (coalesced)
        tile[noff * 33 + c] = img[(size_t)c * NPIX + n0 + noff];
    }
    __syncthreads();
#pragma unroll
    for (int k = 0; k < 4; ++k) {
        int lin = t + k * 256;
        int noff = lin >> 5, c = lin & 31;      // wave: fixed noff, c 0..31 (coalesced)
        outp[(size_t)(n0 + noff) * CH + c] = tile[noff * 33 + c];
    }
}

// ---------------------------------------------------------------------------
// Kernel 1b: fold the uniform t interpolation. For the two selected time
// planes, produce col[plane][c][y] = wx0*P[c,y,x0] + wx1*P[c,y,x1].
// Only 2 * 32 * 512 floats = 128 KB: the entire time-plane working set.
// ---------------------------------------------------------------------------
__global__ __launch_bounds__(256) void pack_timecols(const float* __restrict__ tsp,
                                                     const int* __restrict__ idxp,
                                                     const float* __restrict__ t_emb,
                                                     float* __restrict__ colws) {
    float tv  = t_emb[0];
    float ix  = (tv + 1.0f) * 0.5f * (float)(TT - 1);
    float ix0f = floorf(ix);
    float wx1 = ix - ix0f, wx0 = ix0f + 1.0f - ix;   // weights from UNclamped floor (as ref)
    int x0 = iclamp((int)ix0f, 0, TT - 1);
    int x1 = iclamp((int)ix0f + 1, 0, TT - 1);
    int idx = *idxp;

    int gid = blockIdx.x * blockDim.x + threadIdx.x;  // over 2*CH*RR = 32768
    if (gid >= COL_ELEMS) return;
    int plane = gid >> 14;             // / (CH*RR)
    int rem   = gid & (CH * RR - 1);   // c*512 + y
    int c = rem >> 9, y = rem & (RR - 1);
    const float* img = tsp + ((size_t)(2 * idx + plane) * CH + c) * (size_t)(RR * TT);
    colws[gid] = wx0 * img[y * TT + x0] + wx1 * img[y * TT + x1];
}

// ---------------------------------------------------------------------------
// Kernel 2: main sampler. Stage the 128 KB column tables into LDS (async-to-
// LDS path on CDNA5 when available), then for each pixel:
//   - space plane: 4 corner blocks, each one contiguous 128B (32ch) line,
//     read as 8x float4 per corner, bilinear-weighted.
//   - time planes: 2 ds_load_b32 per channel per plane from LDS.
// Output streamed with non-temporal stores (write-once 402 MB; keep L2 for
// the gather-resident planes).
// ---------------------------------------------------------------------------
__global__ __launch_bounds__(256) void sample_kernel(const float* __restrict__ samples,
                                                     const float* __restrict__ sp_packed,
                                                     const float* __restrict__ colws,
                                                     float* __restrict__ outp) {
    __shared__ float col[COL_ELEMS];   // 128 KB of 320 KB/WGP
    const int tid = threadIdx.x;

#if defined(__has_builtin) && __has_builtin(__builtin_amdgcn_global_load_async_to_lds_b128)
    {
        typedef __attribute__((address_space(3))) float  lds_f;
        typedef __attribute__((address_space(1))) float  glb_f;
        typedef __attribute__((address_space(1))) v4i    v4i_g;   // "int4 __device__*"
        typedef __attribute__((address_space(3))) v4i    v4i_l;   // "int4 __shared__*"
        // AS(3) pointers are 32-bit LDS byte offsets; generic LDS ptr low bits hold it.
        lds_f* col3 = (lds_f*)(uint32_t)(uintptr_t)&col[0];
        glb_f* g    = (glb_f*)colws;
        for (int i = tid * 4; i < COL_ELEMS; i += 256 * 4) {
            __builtin_amdgcn_global_load_async_to_lds_b128(
                (v4i_g*)(g + i), (v4i_l*)(col3 + i), 0, 0);
        }
        __builtin_amdgcn_s_wait_asynccnt(0);
    }
#else
    for (int i = tid * 4; i < COL_ELEMS; i += 256 * 4)
        *(v4f*)&col[i] = *(const v4f*)&colws[i];
#endif
    __syncthreads();

    const int stride = gridDim.x * 256;
    for (int p = blockIdx.x * 256 + tid; p < HW; p += stride) {
        float sx = samples[2 * p + 0];
        float sy = samples[2 * p + 1];

        float ix = (sx + 1.0f) * 0.5f * 511.0f;   // also iy of time-plane 0
        float iy = (sy + 1.0f) * 0.5f * 511.0f;   // also iy of time-plane 1
        float ix0f = floorf(ix), iy0f = floorf(iy);
        float wx1 = ix - ix0f, wx0 = ix0f + 1.0f - ix;
        float wy1 = iy - iy0f, wy0 = iy0f + 1.0f - iy;
        int x0 = iclamp((int)ix0f, 0, 511);
        int x1 = iclamp((int)ix0f + 1, 0, 511);
        int y0 = iclamp((int)iy0f, 0, 511);
        int y1 = iclamp((int)iy0f + 1, 0, 511);

        float w_nw = wx0 * wy0, w_ne = wx1 * wy0;
        float w_sw = wx0 * wy1, w_se = wx1 * wy1;

        const v4f* pnw = (const v4f*)(sp_packed + (size_t)((y0 << 9) + x0) * CH);
        const v4f* pne = (const v4f*)(sp_packed + (size_t)((y0 << 9) + x1) * CH);
        const v4f* psw = (const v4f*)(sp_packed + (size_t)((y1 << 9) + x0) * CH);
        const v4f* pse = (const v4f*)(sp_packed + (size_t)((y1 << 9) + x1) * CH);

        float* o0 = outp + p;
#pragma unroll
        for (int cq = 0; cq < CH / 4; ++cq) {
            v4f a = pnw[cq], b = pne[cq], c2 = psw[cq], d = pse[cq];
#pragma unroll
            for (int j = 0; j < 4; ++j) {
                int c = cq * 4 + j;
                float fsp = a[j] * w_nw + b[j] * w_ne + c2[j] * w_sw + d[j] * w_se;
                float ft0 = wx0 * col[(c << 9) + x0] + wx1 * col[(c << 9) + x1];
                float ft1 = wy0 * col[(CH * RR) + (c << 9) + y0]
                          + wy1 * col[(CH * RR) + (c << 9) + y1];
                __builtin_nontemporal_store(fsp, o0 + (size_t)c * HW);
                __builtin_nontemporal_store(ft0, o0 + (size_t)(CH + c) * HW);
                __builtin_nontemporal_store(ft1, o0 + (size_t)(2 * CH + c) * HW);
            }
        }
    }
}

// ---------------------------------------------------------------------------
extern "C" void kernel_launch(void* const* d_in, const int* in_sizes, int n_in,
                              void* d_out, int out_size, void* d_ws, size_t ws_size,
                              hipStream_t stream) {
    const float* samples = (const float*)d_in[0];  // (1,1024,1024,2)
    const float* t_emb   = (const float*)d_in[1];  // (1,)
    const float* sp_all  = (const float*)d_in[2];  // (3,32,512,512)
    const float* tsp     = (const float*)d_in[3];  // (6,32,512,128)
    const int*   idxp    = (const int*)d_in[4];    // scalar
    float*       outp    = (float*)d_out;          // (96, 1M)

    float* sp_packed = (float*)d_ws;               // 33.5 MB channels-last space plane
    float* colws     = sp_packed + SP_ELEMS;       // 128 KB folded time-plane columns

    pack_space   <<<NPIX / 32,          256, 0, stream>>>(sp_all, idxp, sp_packed);
    pack_timecols<<<COL_ELEMS / 256,    256, 0, stream>>>(tsp, idxp, t_emb, colws);
    sample_kernel<<<1024,               256, 0, stream>>>(samples, sp_packed, colws, outp);
}